// Avg_24129126269602
// MI455X (gfx1250) — compile-verified
//
#include <hip/hip_runtime.h>
#include <hip/hip_bf16.h>

// Problem constants from the reference (B taken from in_sizes at launch).
#define SEQ_L 2048
#define SEQ_D 1024

#define WAVES  8    // waves per block (256 threads, wave32)
#define PDEPTH 8    // async pipeline depth: 8 slots x 512B per wave

typedef float v2f __attribute__((ext_vector_type(2)));
typedef float v8f __attribute__((ext_vector_type(8)));

// ---------------------------------------------------------------------------
// Stage 1: masked row-sum of seq[b, r0:r1, :] for one (batch, L-split) pair.
// Each wave32 owns a 16-column tile. Data is staged 8 rows (512 B) at a time
// with GLOBAL_LOAD_ASYNC_TO_LDS_B128 into a wave-private 8-slot LDS ring
// (ASYNCcnt-tracked, no block barriers), then fed to v_wmma_f32_16x16x4_f32
// with A = ones (row 0 of D = column sums; invariant to B K-permutation).
// Ragged edges (<=7 rows) are added with plain VALU loads.
// ---------------------------------------------------------------------------
__global__ __launch_bounds__(256)
void seg_sum_wmma(const float* __restrict__ seq,
                  const int* __restrict__ begin,
                  const int* __restrict__ end,
                  float* __restrict__ out,   // [S, B, D] partials or [B, D] final
                  int S, int divide) {
    // 8 waves * 8 slots * 128 floats (8 rows x 16 cols) = 32 KB
    __shared__ __align__(16) float smem[WAVES * PDEPTH * 128];

    const int lane = threadIdx.x & 31;
    const int wave = threadIdx.x >> 5;
    const int n    = lane & 15;
    const int col0 = blockIdx.x * 128 + wave * 16;
    const int b    = blockIdx.y;
    const int s    = blockIdx.z;

    const int rb   = begin[b];
    const int re   = end[b];
    const int span = re - rb;

    // Contiguous sub-range of [rb, re) for split s (span*S <= 16384, fits int).
    const int r0 = rb + (span * s) / S;
    const int r1 = rb + (span * (s + 1)) / S;

    const float* __restrict__ base = seq + (size_t)b * SEQ_L * SEQ_D;

    const int nRows = r1 - r0;
    const int n8    = nRows >> 3;          // full 8-row slots via async+WMMA
    const int ib    = r0 + (n8 << 3);      // start of scalar tail

    float* swave = smem + wave * (PDEPTH * 128);

    // Per-lane staging geometry: 32 lanes x 16 B = 8 rows x 64 B (16 cols).
    const int srow = lane >> 2;            // 0..7 : row within slot
    const int scol = (lane & 3) * 4;       // 0,4,8,12 : col within 16-col tile
    const float* gsrc0 = base + (size_t)(r0 + srow) * SEQ_D + col0 + scol;
    // Low 32 bits of a flat shared pointer = workgroup-relative LDS byte offset.
    const unsigned ldsLane =
        (unsigned)(size_t)(swave) + (unsigned)((srow * 16 + scol) * 4);

    v8f acc = {};
    const v2f aOnes = {1.0f, 1.0f};
    const int kA = (lane < 16) ? 0 : 2;    // which K-pair this lane's B carries

    auto stage = [&](int i) {
        const unsigned dst = ldsLane + (unsigned)((i & (PDEPTH - 1)) * 512);
        const unsigned long long g =
            (unsigned long long)(size_t)(gsrc0 + ((size_t)i << 3) * SEQ_D);
        asm volatile("global_load_async_to_lds_b128 %0, %1, off th:TH_LOAD_NT"
                     :: "v"(dst), "v"(g) : "memory");
    };

    auto consume = [&](int c) {
        const float* sw = swave + (c & (PDEPTH - 1)) * 128;
#pragma unroll
        for (int h = 0; h < 2; ++h) {
            v2f bv;
            bv.x = sw[(4 * h + kA) * 16 + n];
            bv.y = sw[(4 * h + kA + 1) * 16 + n];
            acc = __builtin_amdgcn_wmma_f32_16x16x4_f32(
                false, aOnes, false, bv, (short)0, acc, false, false);
        }
    };

    // ---- software pipeline: prologue / steady / drain --------------------
    const int pre = (n8 < PDEPTH) ? n8 : PDEPTH;
    for (int i = 0; i < pre; ++i) stage(i);

    int c = 0;
    for (; c < n8 - pre; ++c) {            // steady state (only if n8 > PDEPTH)
        asm volatile("s_wait_asynccnt 0x7" ::: "memory");  // slot c landed
        consume(c);
        asm volatile("s_wait_dscnt 0x0" ::: "memory");     // reads retired
        stage(c + PDEPTH);                 // reuse slot (c mod PDEPTH)
    }
    asm volatile("s_wait_asynccnt 0x0" ::: "memory");
    for (; c < n8; ++c) consume(c);

    // Ragged tail: up to 7 rows, plain adds on lanes 0..15 (N = lane).
    float eacc = 0.0f;
    for (int l = ib; l < r1; ++l) {
        if (lane < 16) {
            eacc += __builtin_nontemporal_load(base + (size_t)l * SEQ_D + col0 + n);
        }
    }

    if (lane < 16) {
        // C/D layout: VGPR0, lanes 0..15 -> M=0, N=lane : the column sums.
        float r = acc[0] + eacc;
        if (divide) r *= (1.0f / (float)span);
        out[((size_t)s * gridDim.y + b) * SEQ_D + col0 + lane] = r;
    }
}

// ---------------------------------------------------------------------------
// Stage 2: reduce S partials per (b, d) and divide by the segment length.
// One block per batch, 256 threads x float4 = 1024 columns.
// ---------------------------------------------------------------------------
__global__ __launch_bounds__(256)
void seg_finalize(const float* __restrict__ part,  // [S, B, D]
                  const int* __restrict__ begin,
                  const int* __restrict__ end,
                  float* __restrict__ out,         // [B, D]
                  int S, int Bn) {
    const int b = blockIdx.x;
    const int d = threadIdx.x * 4;

    float4 acc = make_float4(0.f, 0.f, 0.f, 0.f);
    for (int s = 0; s < S; ++s) {
        const float4 p = *(const float4*)(part + ((size_t)s * Bn + b) * SEQ_D + d);
        acc.x += p.x; acc.y += p.y; acc.z += p.z; acc.w += p.w;
    }
    const float inv = 1.0f / (float)(end[b] - begin[b]);
    float4 r = make_float4(acc.x * inv, acc.y * inv, acc.z * inv, acc.w * inv);
    *(float4*)(out + (size_t)b * SEQ_D + d) = r;
}

extern "C" void kernel_launch(void* const* d_in, const int* in_sizes, int n_in,
                              void* d_out, int out_size, void* d_ws, size_t ws_size,
                              hipStream_t stream) {
    const float* seq  = (const float*)d_in[0];
    const int*   beg  = (const int*)d_in[1];
    const int*   endp = (const int*)d_in[2];
    float*       out  = (float*)d_out;

    const int Bn = in_sizes[1];                       // 64
    const size_t partBytes = (size_t)Bn * SEQ_D * sizeof(float);

    int S = (partBytes > 0) ? (int)(ws_size / partBytes) : 0;
    if (S > 8) S = 8;

    const dim3 block(256);
    if (S >= 2) {
        // Two-stage: split each batch's row range S ways for occupancy,
        // partials in workspace, then reduce + divide.
        dim3 g1(SEQ_D / 128, Bn, S);
        seg_sum_wmma<<<g1, block, 0, stream>>>(seq, beg, endp, (float*)d_ws, S, 0);
        seg_finalize<<<dim3(Bn), block, 0, stream>>>((const float*)d_ws, beg, endp,
                                                     out, S, Bn);
    } else {
        // Workspace too small: single stage straight into d_out with divide fused.
        dim3 g1(SEQ_D / 128, Bn, 1);
        seg_sum_wmma<<<g1, block, 0, stream>>>(seq, beg, endp, out, 1, 1);
    }
}